// DirectedAPPNPPropagation_16947940950524
// MI455X (gfx1250) — compile-verified
//
#include <hip/hip_runtime.h>
#include <hip/hip_bf16.h>
#include <stdint.h>

#define N_NODES 10000
#define N_EDGES 640000
#define DFEAT   128
#define ALPHA_F 0.1f
#define TILE    256   // edges staged per block; 640000 / 256 = 2500 exact

typedef int v2i __attribute__((vector_size(2 * sizeof(int))));

#define AS_GLOBAL __attribute__((address_space(1)))
#define AS_LDS    __attribute__((address_space(3)))

// Non-returning hardware f32 atomic add (global_atomic_add_f32, no RTN).
__device__ __forceinline__ void atomic_add_f32(float* p, float v) {
    (void)__hip_atomic_fetch_add(p, v, __ATOMIC_RELAXED, __HIP_MEMORY_SCOPE_AGENT);
}

// ---------------------------------------------------------------------------
// HOT KERNEL (defined first so the disasm snippet shows it).
// Scatter: one block = one 256-edge tile. Edge metadata is staged into LDS
// with gfx1250 async global->LDS copies (ASYNCcnt), then each wave32 owns
// 32 edges: one b128 gather per lane covers a full 128-f32 feature row,
// scaled and scattered with hardware f32 atomics (out table is L2-resident).
// ---------------------------------------------------------------------------
__global__ __launch_bounds__(256) void scatter_kernel(
        const float* __restrict__ x,
        const long long* __restrict__ ei,
        const float* __restrict__ w,
        const float* __restrict__ ws,
        float* __restrict__ out) {
    __shared__ long long s_row[TILE];
    __shared__ long long s_col[TILE];
    __shared__ float     s_w[TILE];

    const int t    = threadIdx.x;
    const int base = blockIdx.x * TILE;

#if defined(__gfx1250__) && \
    __has_builtin(__builtin_amdgcn_global_load_async_to_lds_b64) && \
    __has_builtin(__builtin_amdgcn_global_load_async_to_lds_b32)
    // Async DMA the tile's edge stream into LDS (tracked by ASYNCcnt).
    __builtin_amdgcn_global_load_async_to_lds_b64(
        (AS_GLOBAL v2i*)(ei + base + t),
        (AS_LDS v2i*)(&s_row[t]), 0, 0);
    __builtin_amdgcn_global_load_async_to_lds_b64(
        (AS_GLOBAL v2i*)(ei + N_EDGES + base + t),
        (AS_LDS v2i*)(&s_col[t]), 0, 0);
    __builtin_amdgcn_global_load_async_to_lds_b32(
        (AS_GLOBAL int*)(w + base + t),
        (AS_LDS int*)(&s_w[t]), 0, 0);
#if __has_builtin(__builtin_amdgcn_s_wait_asynccnt)
    __builtin_amdgcn_s_wait_asynccnt(0);
#else
    asm volatile("s_wait_asynccnt 0" ::: "memory");
#endif
#else
    s_row[t] = ei[base + t];
    s_col[t] = ei[N_EDGES + base + t];
    s_w[t]   = w[base + t];
#endif
    __syncthreads();

    const int lane = t & 31;   // wave32
    const int wid  = t >> 5;   // 8 waves per block

    const float4* __restrict__ x4 = (const float4*)x;

    #pragma unroll 4
    for (int j = 0; j < 32; ++j) {
        const int slot = (wid << 5) | j;
        const int r = (int)s_row[slot];           // LDS broadcast reads
        const int c = (int)s_col[slot];
        const float norm =
            (1.0f - ALPHA_F) * s_w[slot] / fmaxf(ws[r], 1.0f);

        // 32 lanes x float4 = the full 128-float feature row in one b128 load.
        float4 v = x4[(size_t)r * (DFEAT / 4) + lane];

        float* o = out + (size_t)c * DFEAT + (lane << 2);
        atomic_add_f32(o + 0, norm * v.x);
        atomic_add_f32(o + 1, norm * v.y);
        atomic_add_f32(o + 2, norm * v.z);
        atomic_add_f32(o + 3, norm * v.w);
    }
}

// ---------------------------------------------------------------------------
// Degree: weighted out-degree per source node (segment_sum over row).
// ws is only 40KB -> L2/L1 resident, atomics are cheap.
// ---------------------------------------------------------------------------
__global__ void degree_kernel(const long long* __restrict__ ei,   // [2*E], row = ei[0:E]
                              const float* __restrict__ w,
                              float* __restrict__ ws) {
    int e = blockIdx.x * blockDim.x + threadIdx.x;
    if (e < N_EDGES) {
        int r = (int)ei[e];
        atomic_add_f32(&ws[r], w[e]);
    }
}

// ---------------------------------------------------------------------------
// out = ALPHA * x   (float4 / global b128 path)
// ---------------------------------------------------------------------------
__global__ void init_out_kernel(const float4* __restrict__ x4,
                                float4* __restrict__ out4) {
    int i = blockIdx.x * blockDim.x + threadIdx.x;
    if (i < N_NODES * (DFEAT / 4)) {
        float4 v = x4[i];
        out4[i] = make_float4(ALPHA_F * v.x, ALPHA_F * v.y,
                              ALPHA_F * v.z, ALPHA_F * v.w);
    }
}

// ---------------------------------------------------------------------------
// Zero the per-source-node weight accumulator (lives in d_ws).
// ---------------------------------------------------------------------------
__global__ void zero_ws_kernel(float* __restrict__ ws) {
    int i = blockIdx.x * blockDim.x + threadIdx.x;
    if (i < N_NODES) ws[i] = 0.0f;
}

// ---------------------------------------------------------------------------
extern "C" void kernel_launch(void* const* d_in, const int* in_sizes, int n_in,
                              void* d_out, int out_size, void* d_ws, size_t ws_size,
                              hipStream_t stream) {
    (void)in_sizes; (void)n_in; (void)out_size; (void)ws_size;

    const float*     x  = (const float*)d_in[0];      // [N, 128] f32
    const long long* ei = (const long long*)d_in[1];  // [2, E] i64 (row-major)
    const float*     w  = (const float*)d_in[2];      // [E] f32
    float* out = (float*)d_out;                       // [N, 128] f32
    float* ws  = (float*)d_ws;                        // N_NODES floats (40KB)

    zero_ws_kernel<<<(N_NODES + 255) / 256, 256, 0, stream>>>(ws);
    init_out_kernel<<<(N_NODES * (DFEAT / 4) + 255) / 256, 256, 0, stream>>>(
        (const float4*)x, (float4*)out);
    degree_kernel<<<(N_EDGES + 255) / 256, 256, 0, stream>>>(ei, w, ws);
    scatter_kernel<<<N_EDGES / TILE, 256, 0, stream>>>(x, ei, w, ws, out);
}